// HarmonicOscillatorWithInteractionD_89137751261219
// MI455X (gfx1250) — compile-verified
//
#include <hip/hip_runtime.h>
#include <math.h>

// ---------------------------------------------------------------------------
// HarmonicOscillatorWithInteraction local energy, fused CDNA5 kernel.
//   NW=4096 walkers, N=48 coords, H=256 hidden.
//   Dominant cost: per-walker tangent GEMM Z = T(48x256) @ W2(256x256),
//   done with V_WMMA_F32_16X16X4_F32 on W2 staged in 256KB of LDS.
//   W2 is stored row-pair interleaved in LDS so each WMMA B-fragment
//   (even/odd row pair, same column) is one aligned ds_load_b64.
// ---------------------------------------------------------------------------

typedef __attribute__((ext_vector_type(2))) float v2f;
typedef __attribute__((ext_vector_type(8))) float v8f;

#define NWALK 4096
#define NC    48      // coords per walker
#define HID   256     // hidden width
#define WPB   8       // walkers per workgroup
#define BLOCK 256     // 8 waves

#define GAUSS_C 0.79788456080286535588f   // 1/(sqrt(2*pi)*0.5)

// interleaved W2 index: element (i,j) lives at (i>>1)*512 + 2*j + (i&1)
__device__ __forceinline__ int w2i(int i, int j) {
    return ((i >> 1) << 9) + (j << 1) + (i & 1);
}

__global__ __launch_bounds__(BLOCK)
void vmc_local_energy_kernel(const float* __restrict__ x,    // (NW, 16, 3)
                             const float* __restrict__ W1,   // (48, 256)
                             const float* __restrict__ b1,   // (256)
                             const float* __restrict__ W2,   // (256, 256)
                             const float* __restrict__ b2,   // (256)
                             const float* __restrict__ w3,   // (256)
                             const float* __restrict__ b3,   // scalar (unused by grads)
                             float* __restrict__ out)        // eloc|kin|pot|inter, each NW
{
    // ---- LDS (~306 KB, fits CDNA5 320 KB/WGP) ----
    __shared__ float sW2[HID * HID];       // 256 KB: W2, row-pair interleaved
    __shared__ float sX [WPB * NC];        // walker coords
    __shared__ float sC [HID];             // c_i = sum_k W1[k,i]^2
    __shared__ float sH1[WPB * HID];
    __shared__ float sD1[WPB * HID];       // 1 - h1^2
    __shared__ float sS2[WPB * HID];       // w3*(1-h2^2)
    __shared__ float sQ [WPB * HID];       // w3*h2*(1-h2^2)
    __shared__ float sS1[WPB * HID];       // g1 * d1
    __shared__ float sLapA[WPB], sTermB[WPB], sG2[WPB], sPot[WPB], sInter[WPB];

    const int tid  = threadIdx.x;
    const int lane = tid & 31;
    const int wave = tid >> 5;            // 0..7 : walker owned in WMMA phase
    const int wg0  = blockIdx.x * WPB;    // first walker of this WG

    // ------- Phase 0: stage W2 (interleaved), x, c; zero accumulators ------
    {
        // Each iteration handles a 2x2 block: rows (2ip, 2ip+1), cols (2jp, 2jp+1).
        // Interleaved destination is 4 consecutive floats -> one ds_store_b128.
        for (int idx = tid; idx < 128 * 128; idx += BLOCK) {
            int ip = idx >> 7;            // row pair
            int jp = idx & 127;           // col pair
            int i = ip << 1, j = jp << 1;
            v2f r0 = *(const v2f*)&W2[i * HID + j];         // (i,   j..j+1)
            v2f r1 = *(const v2f*)&W2[(i + 1) * HID + j];   // (i+1, j..j+1)
            float4 v4 = make_float4(r0.x, r1.x, r0.y, r1.y);
            *(float4*)&sW2[(ip << 9) + (j << 1)] = v4;
        }

        for (int i = tid; i < WPB * NC; i += BLOCK) sX[i] = x[wg0 * NC + i];

        // c_i = sum_k W1[k,i]^2  (one i per thread)
        float c = 0.f;
        #pragma unroll
        for (int k = 0; k < NC; ++k) { float v = W1[k * HID + tid]; c += v * v; }
        sC[tid] = c;

        if (tid < WPB) { sLapA[tid] = 0.f; sTermB[tid] = 0.f; sG2[tid] = 0.f; }
    }
    __syncthreads();

    // ------------------- Phase 1: layer 1 forward + pot/inter --------------
    {
        float a[WPB];
        float bb = b1[tid];
        #pragma unroll
        for (int w = 0; w < WPB; ++w) a[w] = bb;
        #pragma unroll 4
        for (int k = 0; k < NC; ++k) {
            float wk = W1[k * HID + tid];           // coalesced, L2/WGP$ hot
            #pragma unroll
            for (int w = 0; w < WPB; ++w) a[w] += sX[w * NC + k] * wk;
        }
        #pragma unroll
        for (int w = 0; w < WPB; ++w) {
            float h = tanhf(a[w]);
            sH1[w * HID + tid] = h;
            sD1[w * HID + tid] = 1.f - h * h;
        }

        if (tid < WPB) {                            // cheap per-walker scalars
            const float* xw = &sX[tid * NC];
            float p = 0.f;
            #pragma unroll
            for (int k = 0; k < NC; ++k) p += xw[k] * xw[k];
            sPot[tid] = 0.5f * p;
            // faithful triu((16,3),k=1): pairs (f,a) = (0,1),(0,2),(1,2)
            const int pf[3] = {0, 0, 1}, pa[3] = {1, 2, 2};
            float it = 0.f;
            #pragma unroll
            for (int e = 0; e < 3; ++e) {
                float sq = 0.f;
                #pragma unroll
                for (int b = 0; b < 3; ++b) {
                    float d = xw[pf[e] * 3 + b] - xw[pf[e] * 3 + pa[e]];
                    sq += d * d;
                }
                it += expf(-2.0f * sq);             // exp(-sq/(2*sigma^2)), sigma=0.5
            }
            sInter[tid] = GAUSS_C * it;
        }
    }
    __syncthreads();

    // ------------------- Phase 2: layer 2 forward, s2, q -------------------
    {
        float z[WPB];
        float bb = b2[tid];
        #pragma unroll
        for (int w = 0; w < WPB; ++w) z[w] = bb;
        for (int i = 0; i < HID; ++i) {
            float wv = sW2[w2i(i, tid)];            // bank=(2*tid+(i&1))%64: conflict-free
            #pragma unroll
            for (int w = 0; w < WPB; ++w) z[w] += sH1[w * HID + i] * wv;
        }
        float w3v = w3[tid];
        #pragma unroll
        for (int w = 0; w < WPB; ++w) {
            float h2 = tanhf(z[w]);
            float d2 = 1.f - h2 * h2;
            sS2[w * HID + tid] = w3v * d2;
            sQ [w * HID + tid] = w3v * h2 * d2;
        }
    }
    __syncthreads();

    // --------- Phase 3: g1 = W2 s2 ; s1 = g1*d1 ; lapA += v.g1 -------------
    {
        float g[WPB];
        #pragma unroll
        for (int w = 0; w < WPB; ++w) g[w] = 0.f;
        for (int jj = 0; jj < HID; ++jj) {
            int j = (jj + tid) & (HID - 1);         // stagger -> distinct banks
            float wv = sW2[w2i(tid, j)];
            #pragma unroll
            for (int w = 0; w < WPB; ++w) g[w] += wv * sS2[w * HID + j];
        }
        float c = sC[tid];
        #pragma unroll
        for (int w = 0; w < WPB; ++w) {
            float d1 = sD1[w * HID + tid];
            float h1 = sH1[w * HID + tid];
            sS1[w * HID + tid] = g[w] * d1;
            atomicAdd(&sLapA[w], -2.f * h1 * d1 * c * g[w]);   // v.g1
        }
    }
    __syncthreads();

    // --------- Phase 4: grad_x = W1 s1 ; |grad|^2 --------------------------
    for (int idx = tid; idx < WPB * NC; idx += BLOCK) {
        int w = idx / NC, k = idx - w * NC;
        float gx = 0.f;
        #pragma unroll 4
        for (int i = 0; i < HID; ++i) gx += W1[k * HID + i] * sS1[w * HID + i];
        atomicAdd(&sG2[w], gx * gx);
    }

    // --------- Phase 5: WMMA tangent GEMM  Z = (W1 .* d1) @ W2 -------------
    // One wave per walker. M=48 (3 tiles), N=256 (16 tiles, chunks of 4),
    // K=256 (64 steps of 4).  V_WMMA_F32_16X16X4_F32, f32-exact.
    // A: lane=M, K-pair split across lane halves (VGPR0=K+0, VGPR1=K+1).
    // B: lane=N, same split; interleaved sW2 makes each B frag one b64 load.
    {
        const int w    = wave;
        const int half = lane >> 4;                 // 0/1 : K half of the lane
        const int ln   = lane & 15;                 // A: M index, B/C: N index
        float partial = 0.f;

        for (int nc = 0; nc < 4; ++nc) {
            v8f acc[3][4];
            #pragma unroll
            for (int mt = 0; mt < 3; ++mt)
                #pragma unroll
                for (int t = 0; t < 4; ++t)
                    acc[mt][t] = (v8f)(0.f);

            for (int ks = 0; ks < 64; ++ks) {
                const int kb = ks * 4 + half * 2;   // even: aligned row pair
                // B fragments: one ds_load_b64 each (rows kb,kb+1 at column j)
                v2f bf[4];
                #pragma unroll
                for (int t = 0; t < 4; ++t) {
                    int j = (nc * 4 + t) * 16 + ln;
                    bf[t] = *(const v2f*)&sW2[((kb >> 1) << 9) + (j << 1)];
                }
                const v2f dk = *(const v2f*)&sD1[w * HID + kb];   // b64 broadcast
                #pragma unroll
                for (int mt = 0; mt < 3; ++mt) {
                    int m = mt * 16 + ln;                         // tangent row
                    v2f af = *(const v2f*)&W1[m * HID + kb] * dk; // T = W1 .* d1
                    #pragma unroll
                    for (int t = 0; t < 4; ++t)
                        acc[mt][t] = __builtin_amdgcn_wmma_f32_16x16x4_f32(
                            false, af, false, bf[t], (short)0, acc[mt][t],
                            false, false);
                }
            }
            // epilogue: lapB partial = sum_j q_j * Z^2 (q depends on column only)
            #pragma unroll
            for (int t = 0; t < 4; ++t) {
                int j = (nc * 4 + t) * 16 + ln;
                float qv = sQ[w * HID + j];
                float ss = 0.f;
                #pragma unroll
                for (int mt = 0; mt < 3; ++mt)
                    #pragma unroll
                    for (int r = 0; r < 8; ++r) {
                        float zv = acc[mt][t][r];
                        ss += zv * zv;
                    }
                partial += qv * ss;
            }
        }
        atomicAdd(&sTermB[w], partial);
    }
    __syncthreads();

    // ------------------- Phase 6: finalize ---------------------------------
    if (tid < WPB) {
        int w = tid;
        float lap  = sLapA[w] - 2.f * sTermB[w];
        float kin  = -0.5f * (lap + sG2[w]);
        float pot  = sPot[w];
        float intr = sInter[w];
        int g = wg0 + w;
        out[g]             = kin + pot + intr;   // eloc
        out[NWALK + g]     = kin;
        out[2 * NWALK + g] = pot;
        out[3 * NWALK + g] = intr;
    }
}

extern "C" void kernel_launch(void* const* d_in, const int* in_sizes, int n_in,
                              void* d_out, int out_size, void* d_ws, size_t ws_size,
                              hipStream_t stream) {
    (void)in_sizes; (void)n_in; (void)d_ws; (void)ws_size; (void)out_size;
    const float* x  = (const float*)d_in[0];
    const float* W1 = (const float*)d_in[1];
    const float* b1 = (const float*)d_in[2];
    const float* W2 = (const float*)d_in[3];
    const float* b2 = (const float*)d_in[4];
    const float* w3 = (const float*)d_in[5];
    const float* b3 = (const float*)d_in[6];
    float* out = (float*)d_out;

    dim3 grid(NWALK / WPB);   // 512 workgroups, 8 walkers each
    dim3 block(BLOCK);        // 8 waves of 32
    vmc_local_energy_kernel<<<grid, block, 0, stream>>>(x, W1, b1, W2, b2, w3, b3, out);
}